// GRUNonlinearity_28810640621776
// MI455X (gfx1250) — compile-verified
//
#include <hip/hip_runtime.h>

// GRU scan, B=32, L=131072, H=8 — latency-bound sequential recurrence.
// One workgroup, 2 waves (wave32). Wave w handles batches [16w, 16w+16).
// Per step, per wave:  gh^T(24x16) = Whh(24x8) x h^T(8x16) via
// V_WMMA_F32_16X16X4_F32 (2 M-tiles x 2 K-chunks). A = Whh is loop-invariant
// and preloaded in registers; gi = x*w_ih + b folded into C of the r|z tile.
// Gates use hardware v_tanh_f32. Cross-half moves are full-EXEC ds_swizzle +
// branchless cndmask. The x stream is double-buffered one outer iteration
// ahead so global-load latency never sits on the recurrence chain.

typedef __attribute__((ext_vector_type(2))) float v2f;
typedef __attribute__((ext_vector_type(8))) float v8f;

#define GRU_B 32
#define GRU_L 131072
#define GRU_H 8

__device__ __forceinline__ float tanh_hw(float x) {
  float r;
  asm("v_tanh_f32 %0, %1" : "=v"(r) : "v"(x));
  return r;
}

// Swap lane halves 0-15 <-> 16-31 (ds_swizzle SWAPX16: xor=0x10, and=0x1f).
// Must be executed with all lanes active.
__device__ __forceinline__ float swap16(float x) {
  return __int_as_float(__builtin_amdgcn_ds_swizzle(__float_as_int(x), 0x401f));
}

// Build WMMA B operands (h^T, 8x16) from h[k] (valid in lanes 0-15).
// B-chunk c, VGPR0 = {K=4c lanes0-15, K=4c+2 lanes16-31}, VGPR1 = {4c+1, 4c+3}.
// Swizzles run on all lanes; the half-select is a branchless cndmask.
__device__ __forceinline__ void build_B(const float h[8], bool hi,
                                        v2f& B0, v2f& B1) {
  float s2 = swap16(h[2]);
  float s3 = swap16(h[3]);
  float s6 = swap16(h[6]);
  float s7 = swap16(h[7]);
  B0.x = hi ? s2 : h[0];
  B0.y = hi ? s3 : h[1];
  B1.x = hi ? s6 : h[4];
  B1.y = hi ? s7 : h[5];
}

__global__ __launch_bounds__(64)
void gru_scan_kernel(const float* __restrict__ x,
                     const float* __restrict__ hidden,
                     const float* __restrict__ w_ih,    // (24,1)
                     const float* __restrict__ w_hh,    // (24,8) row-major
                     const float* __restrict__ b_ih,    // (24,)
                     const float* __restrict__ b_hh,    // (24,)
                     const float* __restrict__ head_w,  // (1,8)
                     const float* __restrict__ head_b,  // (1,)
                     float* __restrict__ out)           // (32, L)
{
  const int  lane  = threadIdx.x & 31;
  const int  wave  = threadIdx.x >> 5;     // 0 or 1
  const int  m     = lane & 15;            // batch within the wave's tile
  const bool hi    = lane >= 16;           // upper lane half
  const int  batch = wave * 16 + m;

  // ---- loop-invariant WMMA A operands: tiles of Whh ------------------
  // A 16x4 f32 layout: both lane halves hold M = lane&15;
  // VGPR0 holds K = base+2*hi, VGPR1 holds K = base+2*hi+1.
  v2f a[2][2];
#pragma unroll
  for (int t = 0; t < 2; ++t) {
#pragma unroll
    for (int c = 0; c < 2; ++c) {
      int row = t * 16 + m;                // gate row (tile1 rows >= 24 invalid)
      int col = 4 * c + (hi ? 2 : 0);
      float vx = (row < 24) ? w_hh[row * 8 + col]     : 0.0f;
      float vy = (row < 24) ? w_hh[row * 8 + col + 1] : 0.0f;
      a[t][c].x = vx;
      a[t][c].y = vy;
    }
  }

  // ---- per-lane-half gi coefficients for the r|z tile ----------------
  float wihz[8], bz[8];
#pragma unroll
  for (int k = 0; k < 8; ++k) {
    int j = hi ? (8 + k) : k;              // lower half: r rows, upper: z rows
    wihz[k] = w_ih[j];
    bz[k]   = b_ih[j] + b_hh[j];
  }

  // n-gate input-projection + head coefficients (wave-uniform -> SGPRs).
  float win[8], bin[8], hw[8];
#pragma unroll
  for (int k = 0; k < 8; ++k) {
    win[k] = w_ih[16 + k];
    bin[k] = b_ih[16 + k];
    hw[k]  = head_w[k];
  }
  const float hb = head_b[0];

  // C operand for the n tile: just b_hh[16..23] (r scales only the h part).
  v8f c1;
#pragma unroll
  for (int k = 0; k < 8; ++k) c1[k] = b_hh[16 + k];

  // ---- hidden state: h[k] valid in lanes 0-15 (batch = lane&15) ------
  float h[8];
#pragma unroll
  for (int k = 0; k < 8; ++k) h[k] = hidden[batch * GRU_H + k];

  // B operands carried across steps: built right after each h update so the
  // swizzle latency hides under the head dot / store, off the WMMA chain.
  v2f B0, B1;
  build_B(h, hi, B0, B1);

  const float* xrow = x   + (size_t)batch * GRU_L;
  float*       yrow = out + (size_t)batch * GRU_L;

  // Software-pipelined x stream: xq holds the block being consumed, the next
  // block's load is issued at the top of the iteration and only waited on a
  // full iteration later (scalar-clamped offset, branchless, no OOB).
  float4 xq = *(const float4*)(xrow);

  for (int t0 = 0; t0 < GRU_L; t0 += 4) {
    int tnext = (t0 + 4 < GRU_L) ? (t0 + 4) : 0;   // uniform s_cselect
    float4 xnq = *(const float4*)(xrow + tnext);
    __builtin_prefetch(xrow + t0 + 512, 0, 1);     // global_prefetch_b8, ~2KB ahead
    float xs[4] = {xq.x, xq.y, xq.z, xq.w};
    float ys[4];

#pragma unroll
    for (int u = 0; u < 4; ++u) {
      const float xv = xs[u];

      // C = gi + b_hh for r (lower half) | z (upper half) rows.
      v8f c0;
#pragma unroll
      for (int k = 0; k < 8; ++k) c0[k] = fmaf(xv, wihz[k], bz[k]);

      // Tile0 (rows 0-15: r|z), accumulate over the two K chunks.
      v8f d0 = __builtin_amdgcn_wmma_f32_16x16x4_f32(
          false, a[0][0], false, B0, (short)0, c0, false, false);
      d0 = __builtin_amdgcn_wmma_f32_16x16x4_f32(
          false, a[0][1], false, B1, (short)0, d0, false, false);
      // Tile1 (rows 16-23: n), C = b_hh_n only (r multiplies the h_n part).
      v8f d1 = __builtin_amdgcn_wmma_f32_16x16x4_f32(
          false, a[1][0], false, B0, (short)0, c1, false, false);
      d1 = __builtin_amdgcn_wmma_f32_16x16x4_f32(
          false, a[1][1], false, B1, (short)0, d1, false, false);

      // sigmoid on r|z: both lane halves of d0 are valid gate pre-acts.
      v8f s0;
#pragma unroll
      for (int k = 0; k < 8; ++k)
        s0[k] = fmaf(tanh_hw(0.5f * d0[k]), 0.5f, 0.5f);

      // n = tanh(gi_n + r * (h_n + b_hh_n)); r = s0 lower half, aligned w/ d1.
      float nk[8];
#pragma unroll
      for (int k = 0; k < 8; ++k) {
        float npre = fmaf(s0[k], d1[k], fmaf(xv, win[k], bin[k]));
        nk[k] = tanh_hw(npre);
      }

      // z (upper half of s0) down to lanes 0-15 (full-EXEC swizzle), update h.
      float zl[8];
#pragma unroll
      for (int k = 0; k < 8; ++k) zl[k] = swap16(s0[k]);
#pragma unroll
      for (int k = 0; k < 8; ++k)
        h[k] = fmaf(zl[k], h[k] - nk[k], nk[k]);   // (1-z)*n + z*h

      // Rebuild B for the next step immediately; head dot runs off-chain.
      build_B(h, hi, B0, B1);

      float y = hb;
#pragma unroll
      for (int k = 0; k < 8; ++k) y = fmaf(h[k], hw[k], y);
      ys[u] = y;
    }

    if (lane < 16) {
      float4 yq = make_float4(ys[0], ys[1], ys[2], ys[3]);
      *(float4*)(yrow + t0) = yq;
    }
    xq = xnq;
  }
}

extern "C" void kernel_launch(void* const* d_in, const int* in_sizes, int n_in,
                              void* d_out, int out_size, void* d_ws, size_t ws_size,
                              hipStream_t stream) {
  const float* x      = (const float*)d_in[0];
  const float* hidden = (const float*)d_in[1];
  const float* w_ih   = (const float*)d_in[2];
  const float* w_hh   = (const float*)d_in[3];
  const float* b_ih   = (const float*)d_in[4];
  const float* b_hh   = (const float*)d_in[5];
  const float* head_w = (const float*)d_in[6];
  const float* head_b = (const float*)d_in[7];
  float* out = (float*)d_out;

  // Single workgroup, 2 waves: the scan is one sequential dependency chain
  // per 16-batch tile; everything lives in registers.
  gru_scan_kernel<<<dim3(1), dim3(64), 0, stream>>>(
      x, hidden, w_ih, w_hh, b_ih, b_hh, head_w, head_b, out);
}